// HorizontalLinePool_25623774888170
// MI455X (gfx1250) — compile-verified
//
#include <hip/hip_runtime.h>

// Horizontal line (corner) pooling = inclusive max-scan along W (=128) of an
// NCHW fp32 tensor. Pure bandwidth problem (512 MiB traffic -> ~23 us at
// 23.3 TB/s). One wave32 per row: lane l owns floats [4l, 4l+3].
//
// CDNA5 paths: wave32 shuffle scan + double-buffered
// GLOBAL_LOAD_ASYNC_TO_LDS_B128 pipeline tracked with ASYNCcnt
// (CDNA5 ISA).

namespace {
constexpr int HLP_W      = 128;  // row width (floats)
constexpr int HLP_LANES  = 32;   // wave32
constexpr int HLP_VEC    = 4;    // floats per lane (HLP_W / HLP_LANES)
constexpr int HLP_NWAVES = 8;    // waves per block
constexpr int HLP_BLOCK  = HLP_NWAVES * HLP_LANES;  // 256 threads
}

#if defined(__HIP_DEVICE_COMPILE__) && __has_builtin(__builtin_amdgcn_global_load_async_to_lds_b128)
#define HLP_ASYNC 1

typedef int hlp_v4i __attribute__((ext_vector_type(4)));
typedef hlp_v4i __attribute__((address_space(1)))* hlp_gptr_t;  // global int4*
typedef hlp_v4i __attribute__((address_space(3)))* hlp_lptr_t;  // LDS int4*

#if __has_builtin(__builtin_amdgcn_s_wait_asynccnt)
#define HLP_WAIT_ASYNC(n) __builtin_amdgcn_s_wait_asynccnt(n)
#else
#define HLP_WAIT_ASYNC(n) asm volatile("s_wait_asynccnt %0" ::"n"(n) : "memory")
#endif

__device__ __forceinline__ void hlp_async_load16(const float* gsrc, float* lds) {
    // Reinterpret pointee type in generic AS first, then pure addrspace casts.
    hlp_v4i* gg = reinterpret_cast<hlp_v4i*>(const_cast<float*>(gsrc));
    hlp_v4i* ll = reinterpret_cast<hlp_v4i*>(lds);
    __builtin_amdgcn_global_load_async_to_lds_b128(
        (hlp_gptr_t)gg, (hlp_lptr_t)ll, /*offset=*/0, /*cpol=*/0);
}
#endif

__device__ __forceinline__ float4 hlp_row_cummax(float4 v, int lane) {
    // In-lane serial scan over the 4 owned elements.
    v.y = fmaxf(v.y, v.x);
    v.z = fmaxf(v.z, v.y);
    v.w = fmaxf(v.w, v.z);
    // Wave32 inclusive max-scan of the lane aggregates (5 shuffle steps).
    float agg = v.w;
#pragma unroll
    for (int d = 1; d < HLP_LANES; d <<= 1) {
        float t = __shfl_up(agg, d, HLP_LANES);
        if (lane >= d) agg = fmaxf(agg, t);
    }
    // Exclusive prefix for this lane = inclusive scan of lane-1.
    float pre = __shfl_up(agg, 1, HLP_LANES);
    if (lane > 0) {
        v.x = fmaxf(v.x, pre);
        v.y = fmaxf(v.y, pre);
        v.z = fmaxf(v.z, pre);
        v.w = fmaxf(v.w, pre);
    }
    return v;
}

__global__ __launch_bounds__(HLP_BLOCK) void HorizontalLinePool_25623774888170_kernel(
    const float* __restrict__ x, float* __restrict__ out, int nrows) {
    const int lane = threadIdx.x & (HLP_LANES - 1);
    const int wave = threadIdx.x >> 5;
    const int totalWaves = gridDim.x * HLP_NWAVES;
    const long long row0 = (long long)blockIdx.x * HLP_NWAVES + wave;

#if defined(HLP_ASYNC)
    // Double-buffered async Global->LDS staging: 16 B per lane per row,
    // 512 B per wave per stage. ASYNCcnt completes in-order for loads, so
    // waiting for <=1 outstanding guarantees the *current* row is in LDS.
    __shared__ __align__(16) float smem[2][HLP_NWAVES * HLP_W];
    float* lds0 = &smem[0][wave * HLP_W + lane * HLP_VEC];
    float* lds1 = &smem[1][wave * HLP_W + lane * HLP_VEC];

    if (row0 < nrows) {
        hlp_async_load16(x + row0 * HLP_W + lane * HLP_VEC, lds0);
    }
    int buf = 0;
    for (long long row = row0; row < nrows; row += totalWaves) {
        const long long nxt = row + totalWaves;
        if (nxt < nrows) {
            hlp_async_load16(x + nxt * HLP_W + lane * HLP_VEC, buf ? lds0 : lds1);
            HLP_WAIT_ASYNC(1);   // current row's load (older) has completed
        } else {
            HLP_WAIT_ASYNC(0);
        }
        float4 v = *(const float4*)(buf ? lds1 : lds0);
        v = hlp_row_cummax(v, lane);
        *(float4*)(out + row * HLP_W + lane * HLP_VEC) = v;
        buf ^= 1;
    }
#else
    // Fallback: direct coalesced b128 loads; occupancy hides latency.
    for (long long row = row0; row < nrows; row += totalWaves) {
        float4 v = *(const float4*)(x + row * HLP_W + lane * HLP_VEC);
        v = hlp_row_cummax(v, lane);
        *(float4*)(out + row * HLP_W + lane * HLP_VEC) = v;
    }
#endif
}

extern "C" void kernel_launch(void* const* d_in, const int* in_sizes, int n_in,
                              void* d_out, int out_size, void* d_ws, size_t ws_size,
                              hipStream_t stream) {
    (void)n_in; (void)out_size; (void)d_ws; (void)ws_size;
    const float* x = (const float*)d_in[0];
    float* out = (float*)d_out;

    const int nrows = in_sizes[0] / HLP_W;  // 16*256*128 = 524288 rows

    // ~8 rows per wave so the async double-buffer has work to overlap;
    // 8192 blocks * 8 waves = 65536 waves in flight.
    const int rowsPerWave = 8;
    int waves = (nrows + rowsPerWave - 1) / rowsPerWave;
    int blocks = (waves + HLP_NWAVES - 1) / HLP_NWAVES;
    if (blocks < 1) blocks = 1;

    HorizontalLinePool_25623774888170_kernel<<<blocks, HLP_BLOCK, 0, stream>>>(x, out, nrows);
}